// SpatialTokenGen_1580547972492
// MI455X (gfx1250) — compile-verified
//
#include <hip/hip_runtime.h>
#include <math.h>

// ---------------------------------------------------------------------------
// SpatialTokenGen for MI455X (gfx1250, wave32)
//
// Roofline: 134 MB of x @ 23.3 TB/s ~= 5.8 us floor; ~0.27 GFLOP of math.
// => single-pass stream over x with b128 loads; LN+dot(w1) algebraically
// fused so x is read exactly once. Final 8x4096 x 4096 contraction runs on
// v_wmma_f32_16x16x4_f32 (fp32 WMMA, no precision downgrade).
// ---------------------------------------------------------------------------

typedef float v2f __attribute__((ext_vector_type(2)));
typedef float v8f __attribute__((ext_vector_type(8)));

#define D_FFN   1024
#define SEQ_LEN 4096
#define BATCH   8
#define NROWS   (BATCH * SEQ_LEN)   // 32768

__device__ __forceinline__ float wave_sum32(float v) {
#pragma unroll
    for (int off = 16; off > 0; off >>= 1)
        v += __shfl_xor(v, off, 32);
    return v;
}

// ---------------------------------------------------------------------------
// Pass 1: one wave32 per (b,s) row of x.
//   Computes  S1 = sum x, S2 = sum x^2, Sw = sum x*gamma*w1
//   plus row-constants Cgw = sum gamma*w1, Cbw = sum beta*w1 in the same
//   streamed loop (gamma/beta/w1 are 4KB each -> L0/L2 resident).
//   s[b,s] = rsqrt(var+eps) * (Sw - mu*Cgw) + Cbw + b1
// ---------------------------------------------------------------------------
__global__ __launch_bounds__(256) void stg_rowpass(
    const float* __restrict__ x, const float* __restrict__ gamma,
    const float* __restrict__ beta, const float* __restrict__ w1,
    const float* __restrict__ b1, float* __restrict__ sval)
{
    const int lane = threadIdx.x & 31;
    const int wave = threadIdx.x >> 5;
    const int row  = blockIdx.x * 8 + wave;          // [0, 32768)
    const float* __restrict__ xr = x + (size_t)row * D_FFN;

    // hint the streamer: row is 4KB, prefetch the back half
    __builtin_prefetch(xr + 512, 0, 0);

    float s1 = 0.f, s2 = 0.f, sw = 0.f, cgw = 0.f, cbw = 0.f;

#pragma unroll
    for (int j = 0; j < D_FFN / 128; ++j) {          // 8 iterations
        const int d = j * 128 + lane * 4;            // coalesced float4 per lane
        const float4 xv = *(const float4*)(xr    + d);
        const float4 gv = *(const float4*)(gamma + d);
        const float4 wv = *(const float4*)(w1    + d);
        const float4 bv = *(const float4*)(beta  + d);

        float gw;
        gw = gv.x * wv.x; s1 += xv.x; s2 = fmaf(xv.x, xv.x, s2);
        sw = fmaf(xv.x, gw, sw); cgw += gw; cbw = fmaf(bv.x, wv.x, cbw);
        gw = gv.y * wv.y; s1 += xv.y; s2 = fmaf(xv.y, xv.y, s2);
        sw = fmaf(xv.y, gw, sw); cgw += gw; cbw = fmaf(bv.y, wv.y, cbw);
        gw = gv.z * wv.z; s1 += xv.z; s2 = fmaf(xv.z, xv.z, s2);
        sw = fmaf(xv.z, gw, sw); cgw += gw; cbw = fmaf(bv.z, wv.z, cbw);
        gw = gv.w * wv.w; s1 += xv.w; s2 = fmaf(xv.w, xv.w, s2);
        sw = fmaf(xv.w, gw, sw); cgw += gw; cbw = fmaf(bv.w, wv.w, cbw);
    }

    s1  = wave_sum32(s1);
    s2  = wave_sum32(s2);
    sw  = wave_sum32(sw);
    cgw = wave_sum32(cgw);
    cbw = wave_sum32(cbw);

    if (lane == 0) {
        const float mu  = s1 * (1.f / D_FFN);
        const float var = s2 * (1.f / D_FFN) - mu * mu;
        const float inv = rsqrtf(var + 1e-5f);
        sval[row] = inv * (sw - mu * cgw) + cbw + b1[0];
    }
}

// ---------------------------------------------------------------------------
// Pass 2: c[b] = sum_s s[b,s] * w2[s] via V_WMMA_F32_16X16X4_F32.
//   A (16x4 f32): rows m = batch (m>=8 zero-masked), lane l holds
//     {A[m, 2h], A[m, 2h+1]} with m=l&15, h=l>>4  -> contiguous float2 load.
//   B (4x16 f32): B[k,n] = w2[kbase+k] broadcast over n -> same float2 shape.
//   8 waves split K=4096 into 512-wide slices; 4 accumulator chains each.
//   D[m,0] lands in accumulator element m on lane 0 for m=0..7.
// ---------------------------------------------------------------------------
__global__ __launch_bounds__(256) void stg_finish(
    const float* __restrict__ sval, const float* __restrict__ w2,
    const float* __restrict__ b2, float* __restrict__ out)
{
    __shared__ float lds[8 * 8];

    const int lane = threadIdx.x & 31;
    const int wave = threadIdx.x >> 5;
    const int m    = lane & 15;
    const int h    = lane >> 4;                       // 0 or 1
    const float mMask = (m < BATCH) ? 1.f : 0.f;      // zero rows 8..15, no divergence
    const float* __restrict__ arow = sval + (size_t)(m & 7) * SEQ_LEN;

    v8f acc0 = {}; v8f acc1 = {}; v8f acc2 = {}; v8f acc3 = {};

    const int kbeg = wave * (SEQ_LEN / 8);            // 512-wide slice per wave
#pragma unroll 1
    for (int k = kbeg; k < kbeg + SEQ_LEN / 8; k += 16) {
#define WSTEP(ACC, KO)                                                        \
        {                                                                     \
            const int kb = k + (KO);                                          \
            v2f a = *(const v2f*)(arow + kb + 2 * h);                         \
            a = a * mMask;                                                    \
            const v2f b = *(const v2f*)(w2 + kb + 2 * h);                     \
            ACC = __builtin_amdgcn_wmma_f32_16x16x4_f32(                      \
                false, a, false, b, (short)0, ACC, false, false);             \
        }
        WSTEP(acc0, 0)
        WSTEP(acc1, 4)
        WSTEP(acc2, 8)
        WSTEP(acc3, 12)
#undef WSTEP
    }

    const v8f acc = (acc0 + acc1) + (acc2 + acc3);

    if (lane == 0) {
#pragma unroll
        for (int i = 0; i < 8; ++i) lds[wave * 8 + i] = acc[i];
    }
    __syncthreads();

    if (threadIdx.x == 0) {
        float t = 0.f;
#pragma unroll
        for (int i = 0; i < 64; ++i) t += lds[i];
        const float mean_c = t * (1.f / BATCH) + b2[0];
        out[0] = 1.f / (1.f + expf(-mean_c));
    }
}

// ---------------------------------------------------------------------------
extern "C" void kernel_launch(void* const* d_in, const int* in_sizes, int n_in,
                              void* d_out, int out_size, void* d_ws, size_t ws_size,
                              hipStream_t stream) {
    (void)in_sizes; (void)n_in; (void)out_size; (void)ws_size;
    const float* x     = (const float*)d_in[0];
    const float* gamma = (const float*)d_in[1];
    const float* beta  = (const float*)d_in[2];
    const float* w1    = (const float*)d_in[3];
    const float* b1    = (const float*)d_in[4];
    const float* w2    = (const float*)d_in[5];
    const float* b2    = (const float*)d_in[6];
    float* out  = (float*)d_out;
    float* sval = (float*)d_ws;                       // 32768 floats = 128 KB

    stg_rowpass<<<NROWS / 8, 256, 0, stream>>>(x, gamma, beta, w1, b1, sval);
    stg_finish<<<1, 256, 0, stream>>>(sval, w2, b2, out);
}